// RandomPatchEmbedding_80882824118411
// MI455X (gfx1250) — compile-verified
//
#include <hip/hip_runtime.h>
#include <hip/hip_bf16.h>

typedef float v2f __attribute__((ext_vector_type(2)));
typedef float v4f __attribute__((ext_vector_type(4)));
typedef float v8f __attribute__((ext_vector_type(8)));
typedef int   v4i __attribute__((vector_size(16)));   // matches builtin param type

#define AS_GLOBAL __attribute__((address_space(1)))
#define AS_LDS    __attribute__((address_space(3)))

typedef v4i AS_GLOBAL* v4i_gptr;   // pointer to v4i in global (AS1)
typedef v4i AS_LDS*    v4i_lptr;   // pointer to v4i in LDS (AS3)

#if defined(__AMDGCN__) && __has_builtin(__builtin_amdgcn_global_load_async_to_lds_b128)
#define USE_ASYNC_LDS 1
#else
#define USE_ASYNC_LDS 0
#endif

#if USE_ASYNC_LDS
#if __has_builtin(__builtin_amdgcn_s_wait_asynccnt)
#define WAIT_ASYNCCNT(n) __builtin_amdgcn_s_wait_asynccnt(n)
#else
#define WAIT_ASYNCCNT(n) asm volatile("s_wait_asynccnt %0" ::"n"(n) : "memory")
#endif
#else
#define WAIT_ASYNCCNT(n) ((void)0)
#endif

namespace {
constexpr int kB  = 32;
constexpr int kC  = 3;
constexpr int kH  = 384;
constexpr int kW  = 384;
constexpr int kNP = 576;   // patches per image
constexpr int kED = 768;   // embed dim
constexpr int kPD = 768;   // flattened patch dim (C*16*16)

constexpr int MT  = 64;    // patch rows per workgroup tile
constexpr int NT  = 128;   // output channels per workgroup tile
constexpr int KC  = 64;    // K chunk staged in LDS
constexpr int LDA = KC + 4;  // 68-float stride: %64==4 -> conflict-free, 16B-aligned rows
constexpr int LDB = KC + 4;
constexpr int kChunks = kPD / KC;                 // 12
constexpr int kOpsPerChunk = 12;                  // async b128 ops/thread/chunk (4 A + 8 B)

constexpr int kNTilesN      = kED / NT;           // 6
constexpr int kMTilesPerImg = kNP / MT;           // 9 (576 = 9*64: tiles never cross images)
constexpr int kMTiles       = kB * kMTilesPerImg; // 288
}

// Move one 16-float (64 B) run global -> LDS. Async DMA on CDNA5, sync fallback else.
__device__ __forceinline__ void copy16_to_lds(const float* __restrict__ g, float* l) {
#if USE_ASYNC_LDS
  v4i_gptr gp = (v4i_gptr)(g);
  v4i_lptr lp = (v4i_lptr)(l);
  __builtin_amdgcn_global_load_async_to_lds_b128(gp + 0, lp + 0, 0, 0);
  __builtin_amdgcn_global_load_async_to_lds_b128(gp + 1, lp + 1, 0, 0);
  __builtin_amdgcn_global_load_async_to_lds_b128(gp + 2, lp + 2, 0, 0);
  __builtin_amdgcn_global_load_async_to_lds_b128(gp + 3, lp + 3, 0, 0);
#else
  v4f r0 = *(const v4f*)(g + 0);
  v4f r1 = *(const v4f*)(g + 4);
  v4f r2 = *(const v4f*)(g + 8);
  v4f r3 = *(const v4f*)(g + 12);
  *(v4f*)(l + 0)  = r0;
  *(v4f*)(l + 4)  = r1;
  *(v4f*)(l + 8)  = r2;
  *(v4f*)(l + 12) = r3;
#endif
}

__global__ __launch_bounds__(256) void patch_embed_wmma_f32(
    const float* __restrict__ x,
    const int*   __restrict__ h_idx,
    const int*   __restrict__ w_idx,
    const float* __restrict__ Wm,     // (ED, PD) row-major
    const float* __restrict__ bias,   // (ED)
    float*       __restrict__ out)    // (B*NP, ED)
{
  __shared__ float As[2][MT * LDA];   // double-buffered gathered patch tile, [m][k]
  __shared__ float Bs[2][NT * LDB];   // double-buffered W slice, [n][k]

  const int tid = threadIdx.x;

  const int tile   = blockIdx.x;
  const int n_tile = tile % kNTilesN;
  const int m_tile = tile / kNTilesN;
  const int b_img  = m_tile / kMTilesPerImg;
  const int m_img  = (m_tile % kMTilesPerImg) * MT;   // first patch of this tile
  const int n0     = n_tile * NT;

  // ---- per-thread gather descriptor (invariant across K chunks) ----
  const int a_row = tid >> 2;          // 0..63 : patch row within tile
  const int a_seg = tid & 3;           // 0..3  : 16-float segment of the K chunk
  const int pn = m_img + a_row;
  const int hh = h_idx[b_img * kNP + pn];
  const int ww = w_idx[b_img * kNP + pn];
  // &x[b_img][c=0][hh*16][ww*16]
  const float* xpatch =
      x + (((size_t)b_img * kC) * kH + (size_t)hh * 16) * kW + ww * 16;

  // B-load assignment (two 16-float segments per thread)
  const int b_nr0 = tid >> 2;                  // segment tid
  const int b_ks0 = (tid & 3) * 16;
  const int b_nr1 = (tid + 256) >> 2;          // segment tid+256
  const int b_ks1 = b_ks0;                     // (tid+256)&3 == tid&3

  // ---- wave decomposition: 2x4 waves, each owns a 32x32 sub-tile ----
  const int lane = tid & 31;
  const int wid  = tid >> 5;
  const int wm   = wid >> 2;           // 0..1 : 32-row band
  const int wn   = wid & 3;            // 0..3 : 32-col band
  const int lm   = lane & 15;          // M (A frag) / N (B frag) within 16
  const int lk   = (lane >> 4) << 1;   // K sub-offset: 0 or 2

  v8f acc00 = {}; v8f acc01 = {}; v8f acc10 = {}; v8f acc11 = {};

  // Stage one K-chunk into buffer `buf` (3 x 64B per thread).
  auto stage = [&](int k0, int buf) {
    // A: gathered patch pixels, one pw-run per thread
    const int p  = k0 + a_seg * 16;    // flattened (c, ph, pw=0)
    const int c  = p >> 8;
    const int ph = (p >> 4) & 15;
    copy16_to_lds(xpatch + ((size_t)c * kH + ph) * kW,
                  &As[buf][a_row * LDA + a_seg * 16]);
    // B: W[n0+nr][k0+ks .. +16)
    copy16_to_lds(Wm + (size_t)(n0 + b_nr0) * kPD + k0 + b_ks0,
                  &Bs[buf][b_nr0 * LDB + b_ks0]);
    copy16_to_lds(Wm + (size_t)(n0 + b_nr1) * kPD + k0 + b_ks1,
                  &Bs[buf][b_nr1 * LDB + b_ks1]);
  };

  stage(0, 0);  // prologue: chunk 0 in flight

  for (int ci = 0; ci < kChunks; ++ci) {
    const int cur = ci & 1;
    if (ci + 1 < kChunks) {
      // Buffer cur^1 was last read at iteration ci-1; the barrier that ended
      // that iteration licenses this overwrite.
      stage((ci + 1) * KC, cur ^ 1);
      WAIT_ASYNCCNT(kOpsPerChunk);   // chunk ci landed (async loads retire in order)
    } else {
      WAIT_ASYNCCNT(0);
    }
    __syncthreads();                  // publish every wave's chunk-ci data

    const float* Ac = As[cur];
    const float* Bc = Bs[cur];
    #pragma unroll
    for (int kk = 0; kk < KC; kk += 4) {
      // A frag: lane -> (M=lm, K=kk+lk, kk+lk+1)
      const v2f a0 = *(const v2f*)&Ac[(wm * 32 +      lm) * LDA + kk + lk];
      const v2f a1 = *(const v2f*)&Ac[(wm * 32 + 16 + lm) * LDA + kk + lk];
      // B frag: lane -> (N=lm, K=kk+lk, kk+lk+1); Bs is [n][k] so this is B[k][n]
      const v2f b0 = *(const v2f*)&Bc[(wn * 32 +      lm) * LDB + kk + lk];
      const v2f b1 = *(const v2f*)&Bc[(wn * 32 + 16 + lm) * LDB + kk + lk];
      acc00 = __builtin_amdgcn_wmma_f32_16x16x4_f32(false, a0, false, b0, (short)0, acc00, false, false);
      acc01 = __builtin_amdgcn_wmma_f32_16x16x4_f32(false, a0, false, b1, (short)0, acc01, false, false);
      acc10 = __builtin_amdgcn_wmma_f32_16x16x4_f32(false, a1, false, b0, (short)0, acc10, false, false);
      acc11 = __builtin_amdgcn_wmma_f32_16x16x4_f32(false, a1, false, b1, (short)0, acc11, false, false);
    }
    __syncthreads();                  // all waves done with buffer cur
  }

  // ---- epilogue: bias + store. C layout: VGPR r -> M = r (lanes 0-15) / r+8 (16-31)
  const int row_base = b_img * kNP + m_img + wm * 32 + ((lane >> 4) << 3);
  v8f accs[2][2] = { { acc00, acc01 }, { acc10, acc11 } };
  #pragma unroll
  for (int sm = 0; sm < 2; ++sm) {
    #pragma unroll
    for (int sn = 0; sn < 2; ++sn) {
      const int ncol = n0 + wn * 32 + sn * 16 + lm;
      const float bv = bias[ncol];
      float* op = out + (size_t)(row_base + sm * 16) * kED + ncol;
      v8f a = accs[sm][sn];
      #pragma unroll
      for (int r = 0; r < 8; ++r)
        op[(size_t)r * kED] = a[r] + bv;
    }
  }
}

extern "C" void kernel_launch(void* const* d_in, const int* in_sizes, int n_in,
                              void* d_out, int out_size, void* d_ws, size_t ws_size,
                              hipStream_t stream) {
  const float* x     = (const float*)d_in[0];
  const int*   h_idx = (const int*)d_in[1];
  const int*   w_idx = (const int*)d_in[2];
  const float* Wm    = (const float*)d_in[3];
  const float* bias  = (const float*)d_in[4];
  float*       out   = (float*)d_out;

  dim3 grid(kMTiles * kNTilesN);   // 288 * 6 = 1728 workgroups
  dim3 block(256);                 // 8 wave32 waves
  hipLaunchKernelGGL(patch_embed_wmma_f32, grid, block, 0, stream,
                     x, h_idx, w_idx, Wm, bias, out);
}